// OnlineClustering_26130581028905
// MI455X (gfx1250) — compile-verified
//
#include <hip/hip_runtime.h>
#include <math.h>

// ---------------------------------------------------------------------------
// OnlineClustering (Sinkhorn-Knopp) for MI455X / gfx1250, wave32 + WMMA.
//   B=8, N=1024, D=768, K=4096.  M = B*N = 8192 rows.
//   d_out = [assignments 8192*4096 | loss]  (fp32)
//   d_ws  = [logits L | inv_norm | shift | c | r | rowloss | partials]
// GEMM uses v_wmma_f32_16x16x4_f32 with (if available) double-buffered
// global_load_async_to_lds_b128 staging + s_wait_asynccnt.
// ---------------------------------------------------------------------------

typedef float v2f __attribute__((ext_vector_type(2)));
typedef float v8f __attribute__((ext_vector_type(8)));

#if defined(__has_builtin)
# if __has_builtin(__builtin_amdgcn_global_load_async_to_lds_b128) && \
     __has_builtin(__builtin_amdgcn_s_wait_asynccnt)
#  define OC_ASYNC 1
# endif
#endif
#ifndef OC_ASYNC
# define OC_ASYNC 0
#endif

#if OC_ASYNC
// builtin prototype wants pointers to 16-byte int vectors in AS1 (global) / AS3 (LDS)
typedef int oc_v4i __attribute__((vector_size(16)));
typedef __attribute__((address_space(1))) oc_v4i* oc_gptr;
typedef __attribute__((address_space(3))) oc_v4i* oc_lptr;
#endif

namespace oc {
constexpr int BATCH = 8;
constexpr int NN    = 1024;
constexpr int DD    = 768;
constexpr int KK    = 4096;
constexpr int MM    = BATCH * NN;          // 8192
constexpr int BKK   = BATCH * KK;          // 32768 (b,k) columns
constexpr float TGT_T  = 0.06f;
constexpr float PRD_T  = 0.12f;
constexpr float EPS    = 1e-8f;
constexpr float EXPMAX = 50.0f;

constexpr int BM = 128, BN = 128, BD = 32, LPAD = 36;  // LDS pitch: 16B-aligned, conflict-free
constexpr int NSTAGE = DD / BD;                        // 24
constexpr int SPL = 8;                                  // N-split for column passes
constexpr int CH  = NN / SPL;                           // 128 rows per chunk
}

// ---------------- fixed-order block reductions (256 threads) ----------------
__device__ __forceinline__ float blk_sum(float v, float* s) {
    const int t = threadIdx.x;
    s[t] = v; __syncthreads();
    #pragma unroll
    for (int o = 128; o > 0; o >>= 1) { if (t < o) s[t] += s[t + o]; __syncthreads(); }
    float r = s[0]; __syncthreads();
    return r;
}
__device__ __forceinline__ float blk_max(float v, float* s) {
    const int t = threadIdx.x;
    s[t] = v; __syncthreads();
    #pragma unroll
    for (int o = 128; o > 0; o >>= 1) { if (t < o) s[t] = fmaxf(s[t], s[t + o]); __syncthreads(); }
    float r = s[0]; __syncthreads();
    return r;
}

// ---------------- K1: per-row inverse L2 norm (F.normalize semantics) -------
__global__ __launch_bounds__(256) void k_rownorm(const float* __restrict__ x,
                                                 float* __restrict__ inv_norm) {
    __shared__ float sred[256];
    const int row = blockIdx.x;
    const float* xr = x + (size_t)row * oc::DD;
    float ss = 0.f;
    for (int j = threadIdx.x; j < oc::DD; j += 256) { float v = xr[j]; ss += v * v; }
    float SS = blk_sum(ss, sred);
    if (threadIdx.x == 0) {
        float n = sqrtf(SS);
        inv_norm[row] = 1.0f / fmaxf(n, 1e-7f);
    }
}

// ---------------- K2: r-factor init ----------------------------------------
__global__ void k_init_r(float* __restrict__ r) {
    int g = blockIdx.x * blockDim.x + threadIdx.x;
    if (g < oc::MM) r[g] = 1.0f;
}

// ---------------- tile staging: global -> LDS ------------------------------
__device__ __forceinline__ void stage_tile(const float* __restrict__ g, int rbase, int dt,
                                           int tid, float (*dst)[oc::LPAD]) {
    #pragma unroll
    for (int i = 0; i < 4; ++i) {
        const int idx = tid + i * 256;        // 0..1023 float4 slots (128 rows x 8)
        const int row = idx >> 3;
        const int c4  = idx & 7;
        const float* src = g + (size_t)(rbase + row) * oc::DD + dt + c4 * 4;
#if OC_ASYNC
        // async VGPR-bypassing copy, tracked by ASYNCcnt
        __builtin_amdgcn_global_load_async_to_lds_b128(
            (oc_gptr)src, (oc_lptr)&dst[row][c4 * 4], 0, 0);
#else
        *(float4*)&dst[row][c4 * 4] = *(const float4*)src;
#endif
    }
}

// ---------------- K3: GEMM  L[m,k] = inv_norm[m] * (x[m,:]·W[k,:]) + b[k] ---
__global__ __launch_bounds__(256) void k_gemm_wmma(const float* __restrict__ x,
                                                   const float* __restrict__ W,
                                                   const float* __restrict__ bias,
                                                   const float* __restrict__ inv_norm,
                                                   float* __restrict__ L) {
    using namespace oc;
    __shared__ float As[2][BM][LPAD];
    __shared__ float Ws[2][BN][LPAD];

    const int kbase = blockIdx.x * BN;
    const int mbase = blockIdx.y * BM;
    const int tid  = threadIdx.x;
    const int wave = tid >> 5;
    const int lane = tid & 31;
    const int hi   = lane >> 4;      // half-lane group selects K-pair / M+8
    const int lo   = lane & 15;
    const int mw   = wave * 16;      // wave's 16-row strip within block tile

    v8f acc[8];
    #pragma unroll
    for (int t = 0; t < 8; ++t) {
        #pragma unroll
        for (int j = 0; j < 8; ++j) acc[t][j] = 0.0f;
    }

    // prologue: stage 0 into buffer 0
    stage_tile(x, mbase, 0, tid, As[0]);
    stage_tile(W, kbase, 0, tid, Ws[0]);

    for (int s = 0; s < NSTAGE; ++s) {
#if OC_ASYNC
        __builtin_amdgcn_s_wait_asynccnt(0);   // this wave's stage-s copies landed
#endif
        __syncthreads();                        // all waves' stage-s data in LDS;
                                                // all waves done reading buf[(s+1)&1]
        if (s + 1 < NSTAGE) {                   // issue stage s+1 while computing s
            stage_tile(x, mbase, (s + 1) * BD, tid, As[(s + 1) & 1]);
            stage_tile(W, kbase, (s + 1) * BD, tid, Ws[(s + 1) & 1]);
        }
        float (*A)[LPAD] = As[s & 1];
        float (*Bw)[LPAD] = Ws[s & 1];

        #pragma unroll
        for (int dd = 0; dd < BD; dd += 4) {
            // A frag (16x4 f32): lane lo = row m, hi selects K {0,1} vs {2,3}
            v2f a = *(const v2f*)&A[mw + lo][dd + 2 * hi];
            #pragma unroll
            for (int t = 0; t < 8; ++t) {
                // B frag (4x16 f32): lane lo = col n; B[d][n] = W[kbase+t*16+n][d]
                v2f bf = *(const v2f*)&Bw[t * 16 + lo][dd + 2 * hi];
                acc[t] = __builtin_amdgcn_wmma_f32_16x16x4_f32(
                    false, a, false, bf, (short)0, acc[t], false, false);
            }
        }
    }

    // epilogue: fused row scaling + bias, write logits
    #pragma unroll
    for (int t = 0; t < 8; ++t) {
        const int kg = kbase + t * 16 + lo;
        const float bb = bias[kg];
        #pragma unroll
        for (int v = 0; v < 8; ++v) {
            const int mg = mbase + mw + v + 8 * hi;   // C/D layout: VGPR v -> M=v (+8 for hi half)
            L[(size_t)mg * KK + kg] = acc[t][v] * inv_norm[mg] + bb;
        }
    }
}

// ---------------- column passes, split over N into SPL chunks ---------------
// partial column max over one chunk
__global__ __launch_bounds__(256) void k_colmax_p(const float* __restrict__ L,
                                                  float* __restrict__ part) {
    using namespace oc;
    const int g = blockIdx.x * 256 + threadIdx.x;      // [0, BKK)
    const int p = blockIdx.y;                          // chunk
    const int b = g >> 12;
    const int k = g & (KK - 1);
    const size_t base = (size_t)b * NN * KK + (size_t)p * CH * KK + k;
    float mx = -3.402823466e38f;
    for (int n = 0; n < CH; ++n) mx = fmaxf(mx, L[base + (size_t)n * KK]);
    part[(size_t)p * BKK + g] = mx;
}
// combine: shift = max/0.06
__global__ __launch_bounds__(256) void k_shift_comb(const float* __restrict__ part,
                                                    float* __restrict__ shiftv) {
    using namespace oc;
    const int g = blockIdx.x * 256 + threadIdx.x;
    float mx = part[g];
    #pragma unroll
    for (int p = 1; p < SPL; ++p) mx = fmaxf(mx, part[(size_t)p * BKK + g]);
    shiftv[g] = mx / TGT_T;
}
// E = exp(L/0.06 + 50 - shift) over one chunk, with partial column sum
__global__ __launch_bounds__(256) void k_exp_p(const float* __restrict__ L,
                                               const float* __restrict__ shiftv,
                                               float* __restrict__ E,
                                               float* __restrict__ part) {
    using namespace oc;
    const int g = blockIdx.x * 256 + threadIdx.x;
    const int p = blockIdx.y;
    const int b = g >> 12;
    const int k = g & (KK - 1);
    const float sh = shiftv[g];
    const size_t base = (size_t)b * NN * KK + (size_t)p * CH * KK + k;
    float s = 0.f;
    for (int n = 0; n < CH; ++n) {
        const size_t idx = base + (size_t)n * KK;
        float e = expf(L[idx] / TGT_T + EXPMAX - sh);
        E[idx] = e;
        s += e;
    }
    part[(size_t)p * BKK + g] = s;
}
// combine first column normalize: c = 1/(T0+eps)   (c was 1)
__global__ __launch_bounds__(256) void k_c_init(const float* __restrict__ part,
                                                float* __restrict__ cfac) {
    using namespace oc;
    const int g = blockIdx.x * 256 + threadIdx.x;
    float s = part[g];
    #pragma unroll
    for (int p = 1; p < SPL; ++p) s += part[(size_t)p * BKK + g];
    cfac[g] = 1.0f / (s + EPS);
}
// partial weighted column sum T = sum_n E*r over one chunk
__global__ __launch_bounds__(256) void k_colpass_p(const float* __restrict__ E,
                                                   const float* __restrict__ rfac,
                                                   float* __restrict__ part) {
    using namespace oc;
    const int g = blockIdx.x * 256 + threadIdx.x;
    const int p = blockIdx.y;
    const int b = g >> 12;
    const int k = g & (KK - 1);
    const size_t base = (size_t)b * NN * KK + (size_t)p * CH * KK + k;
    const float* rb = rfac + b * NN + p * CH;
    float s = 0.f;
    for (int n = 0; n < CH; ++n) s += E[base + (size_t)n * KK] * rb[n];
    part[(size_t)p * BKK + g] = s;
}
// combine column normalize: c <- c/(c*T+eps)
__global__ __launch_bounds__(256) void k_c_upd(const float* __restrict__ part,
                                               float* __restrict__ cfac) {
    using namespace oc;
    const int g = blockIdx.x * 256 + threadIdx.x;
    float s = part[g];
    #pragma unroll
    for (int p = 1; p < SPL; ++p) s += part[(size_t)p * BKK + g];
    const float cv = cfac[g];
    cfac[g] = cv / (cv * s + EPS);
}

// ---------------- row pass: U = sum_k E*c ; r <- r/(r*U+eps) ----------------
__global__ __launch_bounds__(256) void k_rowpass(const float* __restrict__ E,
                                                 const float* __restrict__ cfac,
                                                 float* __restrict__ rfac) {
    using namespace oc;
    __shared__ float sred[256];
    const int row = blockIdx.x;            // [0, MM)
    const int b = row >> 10;
    const size_t base = (size_t)row * KK;
    const float* cb = cfac + b * KK;
    float u = 0.f;
    for (int k = threadIdx.x; k < KK; k += 256) u += E[base + k] * cb[k];
    float U = blk_sum(u, sred);
    if (threadIdx.x == 0) {
        float rv = rfac[row];
        rfac[row] = rv / (rv * U + EPS);
    }
}

// ---------------- final row pass: last normalize + assignments + row loss ---
__global__ __launch_bounds__(256) void k_final_row(float* __restrict__ E,
                                                   const float* __restrict__ L,
                                                   const float* __restrict__ cfac,
                                                   const float* __restrict__ rfac,
                                                   float* __restrict__ rowloss) {
    using namespace oc;
    __shared__ float sred[256];
    const int row = blockIdx.x;
    const int b = row >> 10;
    const size_t base = (size_t)row * KK;
    const float* cb = cfac + b * KK;

    float u = 0.f, pmax = -3.402823466e38f;
    for (int k = threadIdx.x; k < KK; k += 256) {
        u += E[base + k] * cb[k];
        pmax = fmaxf(pmax, L[base + k] / PRD_T);
    }
    const float U  = blk_sum(u, sred);
    const float PM = blk_max(pmax, sred);
    const float rv = rfac[row];
    const float rfin = rv / (rv * U + EPS);

    float se = 0.f;
    for (int k = threadIdx.x; k < KK; k += 256)
        se += expf(L[base + k] / PRD_T - PM);
    const float SE  = blk_sum(se, sred);
    const float lse = logf(SE) + PM;

    float lossp = 0.f;
    for (int k = threadIdx.x; k < KK; k += 256) {
        const float tgt = E[base + k] * rfin * cb[k];
        lossp -= tgt * (L[base + k] / PRD_T - lse);
        E[base + k] = tgt;                 // assignments in place
    }
    const float LS = blk_sum(lossp, sred);
    if (threadIdx.x == 0) rowloss[row] = LS;
}

// ---------------- final scalar loss -----------------------------------------
__global__ __launch_bounds__(256) void k_loss_reduce(const float* __restrict__ rowloss,
                                                     float* __restrict__ loss_out) {
    using namespace oc;
    __shared__ float sred[256];
    float s = 0.f;
    for (int i = threadIdx.x; i < MM; i += 256) s += rowloss[i];
    float S = blk_sum(s, sred);
    if (threadIdx.x == 0) loss_out[0] = S / (float)MM;
}

// ---------------------------------------------------------------------------
extern "C" void kernel_launch(void* const* d_in, const int* in_sizes, int n_in,
                              void* d_out, int out_size, void* d_ws, size_t ws_size,
                              hipStream_t stream) {
    using namespace oc;
    (void)in_sizes; (void)n_in; (void)out_size; (void)ws_size;

    const float* x    = (const float*)d_in[0];   // [8,1024,768]
    const float* W    = (const float*)d_in[1];   // [4096,768]
    const float* bias = (const float*)d_in[2];   // [4096]

    float* out = (float*)d_out;
    float* E   = out;                            // [MM*KK] exp matrix -> assignments
    float* loss_out = out + (size_t)MM * KK;     // [1]

    float* ws       = (float*)d_ws;
    float* L        = ws;                        // [MM*KK] logits
    float* inv_norm = L + (size_t)MM * KK;       // [MM]
    float* shiftv   = inv_norm + MM;             // [BKK]
    float* cfac     = shiftv + BKK;              // [BKK]
    float* rfac     = cfac + BKK;                // [MM]
    float* rowloss  = rfac + MM;                 // [MM]
    float* part     = rowloss + MM;              // [SPL*BKK] split partials

    const dim3 colgrid(BKK / 256, SPL);
    const int  combgrid = BKK / 256;

    // 1. inverse row norms of x
    k_rownorm<<<MM, 256, 0, stream>>>(x, inv_norm);
    // 2. r = 1
    k_init_r<<<MM / 256, 256, 0, stream>>>(rfac);
    // 3. WMMA GEMM: logits (double-buffered async LDS staging when available)
    k_gemm_wmma<<<dim3(KK / BN, MM / BM), 256, 0, stream>>>(x, W, bias, inv_norm, L);
    // 4. column max -> shift (split + combine)
    k_colmax_p<<<colgrid, 256, 0, stream>>>(L, part);
    k_shift_comb<<<combgrid, 256, 0, stream>>>(part, shiftv);
    // 5. E = exp(...), Sinkhorn col-normalize #1 (split + combine)
    k_exp_p<<<colgrid, 256, 0, stream>>>(L, shiftv, E, part);
    k_c_init<<<combgrid, 256, 0, stream>>>(part, cfac);
    // 6. row #1
    k_rowpass<<<MM, 256, 0, stream>>>(E, cfac, rfac);
    // 7. col #2
    k_colpass_p<<<colgrid, 256, 0, stream>>>(E, rfac, part);
    k_c_upd<<<combgrid, 256, 0, stream>>>(part, cfac);
    // 8. row #2
    k_rowpass<<<MM, 256, 0, stream>>>(E, cfac, rfac);
    // 9. col #3
    k_colpass_p<<<colgrid, 256, 0, stream>>>(E, rfac, part);
    k_c_upd<<<combgrid, 256, 0, stream>>>(part, cfac);
    // 10. row #3 fused with assignments write + per-row CE loss
    k_final_row<<<MM, 256, 0, stream>>>(E, L, cfac, rfac, rowloss);
    // 11. scalar loss (deterministic tree reduction)
    k_loss_reduce<<<1, 256, 0, stream>>>(rowloss, loss_out);
}